// GroupKANLayer_32719060861485
// MI455X (gfx1250) — compile-verified
//
#include <hip/hip_runtime.h>

typedef float v2f __attribute__((ext_vector_type(2)));
typedef float v8f __attribute__((ext_vector_type(8)));

#define BATCH   131072
#define FEATS   64
#define NGROUP  8
#define GSIZE   8
#define NBINS   16
#define OUTG    64
#define NCHAN   (NGROUP*OUTG)        // 512
#define TILE_M  16
#define NTILES  (BATCH/TILE_M)       // 8192
#define XPITCH  68                   // 16*64 tile padded to avoid bank conflicts
#define BLOCK   256
#define NWAVE   (BLOCK/32)
#define GRID    256                  // 2048 waves -> 4 tiles each (uniform)
#define BN_EPS  1e-5f

// workspace layout (floats): [0,512) sum  [512,1024) sumsq  [1024,1536) scale  [1536,2048) shift

__global__ void zero_stats_kernel(float* ws) {
    int i = blockIdx.x * blockDim.x + threadIdx.x;
    if (i < 2 * NCHAN) ws[i] = 0.0f;
}

__global__ void finalize_kernel(float* ws, const float* __restrict__ gamma,
                                const float* __restrict__ beta) {
    int i = blockIdx.x * blockDim.x + threadIdx.x;
    if (i < NCHAN) {
        const float invB = 1.0f / (float)BATCH;
        float mean = ws[i] * invB;
        float var  = ws[NCHAN + i] * invB - mean * mean;
        float rs   = rsqrtf(var + BN_EPS);
        float sc   = rs * gamma[i];
        ws[2 * NCHAN + i] = sc;
        ws[3 * NCHAN + i] = beta[i] - mean * sc;
    }
}

__device__ __forceinline__ float rbf_eval(float xv, const float* __restrict__ cen,
                                          const float* __restrict__ isg,
                                          const float* __restrict__ w, float lwv) {
    float acc = lwv * xv;
#pragma unroll
    for (int j = 0; j < NBINS; ++j) {
        float d = (xv - cen[j]) * isg[j];
        acc = fmaf(w[j], __expf(-0.5f * d * d), acc);
    }
    return acc;
}

template <bool STATS>
__global__ __launch_bounds__(BLOCK)
void kan_kernel(const float* __restrict__ x,
                const float* __restrict__ centres,
                const float* __restrict__ log_widths,
                const float* __restrict__ rbf_w,
                const float* __restrict__ lin_w,
                const float* __restrict__ proj_w,
                const float* __restrict__ proj_b,
                const long long* __restrict__ indices,
                float* __restrict__ out,
                float* __restrict__ ws) {
    __shared__ float  s_x[NWAVE * TILE_M * XPITCH];
    __shared__ float2 s_B[NGROUP * 4 * 2 * 32];     // pre-swizzled WMMA B operands
    __shared__ float  s_cen[NGROUP * NBINS];
    __shared__ float  s_isg[NGROUP * NBINS];
    __shared__ float  s_w[NGROUP * NBINS];
    __shared__ float  s_lw[NGROUP];
    __shared__ float  s_pb[NCHAN];
    __shared__ int    s_fi[NGROUP * GSIZE];
    __shared__ float  s_extra[2 * NCHAN];           // STATS: sum/sumsq accum; else: scale/shift

    const int tid  = threadIdx.x;
    const int lane = tid & 31;
    const int wib  = tid >> 5;

    // ---- block-level constant staging ----
    for (int i = tid; i < NGROUP * NBINS; i += BLOCK) {
        s_cen[i] = centres[i];
        s_isg[i] = 1.0f / (__expf(log_widths[i]) + 1e-6f);
        s_w[i]   = rbf_w[i];
    }
    if (tid < NGROUP) s_lw[tid] = lin_w[tid];
    for (int i = tid; i < NCHAN; i += BLOCK) s_pb[i] = proj_b[i];
    for (int i = tid; i < NGROUP * GSIZE; i += BLOCK) s_fi[i] = (int)indices[i];
    // B-matrix swizzle: B[k][n] = proj_w[g][tile*16+n][k]
    for (int i = tid; i < NGROUP * 4 * 2 * 32; i += BLOCK) {
        int l = i & 31, q = i >> 5;
        int h = q & 1, t = (q >> 1) & 3, g = q >> 3;
        int hl = l >> 4, n = l & 15;
        int k0 = 4 * h + 2 * hl;
        int base = (g * OUTG + t * 16 + n) * GSIZE + k0;
        s_B[i] = make_float2(proj_w[base], proj_w[base + 1]);
    }
    if constexpr (STATS) {
        for (int i = tid; i < 2 * NCHAN; i += BLOCK) s_extra[i] = 0.0f;
    } else {
        for (int i = tid; i < NCHAN; i += BLOCK) {
            s_extra[i]         = ws[2 * NCHAN + i];   // scale
            s_extra[NCHAN + i] = ws[3 * NCHAN + i];   // shift
        }
    }
    __syncthreads();

    const int totalWaves = GRID * NWAVE;              // 2048; NTILES % totalWaves == 0
    const int gwave = blockIdx.x * NWAVE + wib;
    float* sx = &s_x[wib * TILE_M * XPITCH];

    const int r    = lane & 15;
    const int half = lane >> 4;

    for (int tile = gwave; tile < NTILES; tile += totalWaves) {
        const int rowBase = tile * TILE_M;

        // coalesced stage of the 16x64 x tile into LDS
        for (int e = lane; e < TILE_M * FEATS; e += 32) {
            int rr = e >> 6, ff = e & 63;
            sx[rr * XPITCH + ff] = x[(size_t)(rowBase + rr) * FEATS + ff];
        }
        __syncthreads();

        // per-lane RBF values for the WMMA A operands:
        // lanes 0-15 need k={0,1,4,5}, lanes 16-31 need k={2,3,6,7} of row r
        float4 av[NGROUP];
#pragma unroll
        for (int g = 0; g < NGROUP; ++g) {
            const float* cen = &s_cen[g * NBINS];
            const float* isg = &s_isg[g * NBINS];
            const float* wv  = &s_w[g * NBINS];
            const float  lwv = s_lw[g];
            const int s0 = 2 * half;
            float x0 = sx[r * XPITCH + s_fi[g * GSIZE + s0]];
            float x1 = sx[r * XPITCH + s_fi[g * GSIZE + s0 + 1]];
            float x2 = sx[r * XPITCH + s_fi[g * GSIZE + s0 + 4]];
            float x3 = sx[r * XPITCH + s_fi[g * GSIZE + s0 + 5]];
            av[g].x = rbf_eval(x0, cen, isg, wv, lwv);
            av[g].y = rbf_eval(x1, cen, isg, wv, lwv);
            av[g].z = rbf_eval(x2, cen, isg, wv, lwv);
            av[g].w = rbf_eval(x3, cen, isg, wv, lwv);
        }

        // projection: per (group, n-tile) two chained f32 WMMAs (K = 0..3, 4..7)
#pragma unroll
        for (int g = 0; g < NGROUP; ++g) {
            v2f a0, a1;
            a0[0] = av[g].x; a0[1] = av[g].y;
            a1[0] = av[g].z; a1[1] = av[g].w;
#pragma unroll
            for (int t = 0; t < 4; ++t) {
                const int qb = ((g * 4 + t) * 2) * 32;
                float2 f0 = s_B[qb + lane];
                float2 f1 = s_B[qb + 32 + lane];
                v2f b0, b1;
                b0[0] = f0.x; b0[1] = f0.y;
                b1[0] = f1.x; b1[1] = f1.y;

                v8f c = {0.f, 0.f, 0.f, 0.f, 0.f, 0.f, 0.f, 0.f};
                c = __builtin_amdgcn_wmma_f32_16x16x4_f32(false, a0, false, b0,
                                                          (short)0, c, false, false);
                c = __builtin_amdgcn_wmma_f32_16x16x4_f32(false, a1, false, b1,
                                                          (short)0, c, false, false);

                const int ch   = g * OUTG + t * 16 + (lane & 15);
                const float bias = s_pb[ch];
                if constexpr (STATS) {
                    float sm = 0.0f, sq = 0.0f;
#pragma unroll
                    for (int j = 0; j < 8; ++j) {
                        float v = c[j] + bias;
                        sm += v;
                        sq = fmaf(v, v, sq);
                    }
                    atomicAdd(&s_extra[ch], sm);
                    atomicAdd(&s_extra[NCHAN + ch], sq);
                } else {
                    const float sc = s_extra[ch];
                    const float sh = s_extra[NCHAN + ch];
                    size_t ob = (size_t)(rowBase + 8 * half) * NCHAN + ch;
#pragma unroll
                    for (int j = 0; j < 8; ++j)
                        out[ob + (size_t)j * NCHAN] = fmaf(c[j] + bias, sc, sh);
                }
            }
        }
        __syncthreads();
    }

    if constexpr (STATS) {
        __syncthreads();
        for (int i = tid; i < NCHAN; i += BLOCK) {
            atomicAdd(&ws[i], s_extra[i]);
            atomicAdd(&ws[NCHAN + i], s_extra[NCHAN + i]);
        }
    }
}

extern "C" void kernel_launch(void* const* d_in, const int* in_sizes, int n_in,
                              void* d_out, int out_size, void* d_ws, size_t ws_size,
                              hipStream_t stream) {
    const float*     x   = (const float*)d_in[0];
    const float*     cen = (const float*)d_in[1];
    const float*     lwd = (const float*)d_in[2];
    const float*     rbw = (const float*)d_in[3];
    const float*     lin = (const float*)d_in[4];
    const float*     pw  = (const float*)d_in[5];
    const float*     pb  = (const float*)d_in[6];
    const float*     gam = (const float*)d_in[7];
    const float*     bet = (const float*)d_in[8];
    const long long* idx = (const long long*)d_in[9];
    float* out = (float*)d_out;
    float* ws  = (float*)d_ws;

    zero_stats_kernel<<<4, 256, 0, stream>>>(ws);
    kan_kernel<true><<<GRID, BLOCK, 0, stream>>>(x, cen, lwd, rbw, lin, pw, pb, idx,
                                                 nullptr, ws);
    finalize_kernel<<<2, 256, 0, stream>>>(ws, gam, bet);
    kan_kernel<false><<<GRID, BLOCK, 0, stream>>>(x, cen, lwd, rbw, lin, pw, pb, idx,
                                                  out, ws);
}